// CenterPixelMSE_11424613007985
// MI455X (gfx1250) — compile-verified
//
#include <hip/hip_runtime.h>

// CDNA5 / gfx1250, wave32.
typedef __attribute__((ext_vector_type(2))) float v2f;
typedef __attribute__((ext_vector_type(8))) float v8f;

// Shared wave-wide reduction: one V_WMMA_F32_16X16X4_F32.
//   A(16x4) = {partial, 0} per lane (ISA A-layout: lanes 0-15 -> K0/K1,
//             lanes 16-31 -> K2/K3)  => row sum s[m] = partial[m] + partial[m+16]
//   B(4x16) = all ones (layout-independent)
//   D[m][n] = s[m], replicated across columns.
//   D layout: VGPR r holds M=r (lanes 0-15) / M=r+8 (lanes 16-31), so the sum
//   of the 8 accumulator VGPRs in lane 0 is s[0..7] and in lane 16 is s[8..15].
__device__ __forceinline__ float wave_reduce_wmma(float partial) {
    v2f a;  a.x = partial; a.y = 0.0f;
    v2f bo; bo.x = 1.0f;   bo.y = 1.0f;
    v8f c = {};
    c = __builtin_amdgcn_wmma_f32_16x16x4_f32(
            /*neg_a=*/false, a, /*neg_b=*/false, bo,
            /*c_mod=*/(short)0, c, /*reuse_a=*/false, /*reuse_b=*/false);
    float half8 = ((c[0] + c[1]) + (c[2] + c[3]))
                + ((c[4] + c[5]) + (c[6] + c[7]));
    // Uniform combine of the two half-wave sums (no DS op, no divergence).
    const int lo = __builtin_amdgcn_readlane(__builtin_bit_cast(int, half8), 0);
    const int hi = __builtin_amdgcn_readlane(__builtin_bit_cast(int, half8), 16);
    return __builtin_bit_cast(float, lo) + __builtin_bit_cast(float, hi);
}

// Fast path: ITERS known at compile time -> explicit 3-phase schedule.
// Phase 1: issue ALL coalesced centers/target loads (kept in flight together).
// Phase 2: compute ALL gather addresses, issue ALL pred gathers in flight.
// Phase 3: accumulate. Exactly two dependent memory round trips total.
template <int ITERS>
__global__ __launch_bounds__(32)
void center_mse_wmma_kernel(const float* __restrict__ pred,
                            const float* __restrict__ target,
                            const int2* __restrict__ c2,
                            float* __restrict__ out,
                            int HW, int W, float inv_B)
{
    const int lane = (int)threadIdx.x;          // 0..31, full wave active

    int2  cc[ITERS];
    float tg[ITERS];
#pragma unroll
    for (int i = 0; i < ITERS; ++i) cc[i] = c2[lane + (i << 5)];      // 8B coalesced
#pragma unroll
    for (int i = 0; i < ITERS; ++i) tg[i] = target[lane + (i << 5)];  // 4B coalesced

    float p[ITERS];
#pragma unroll
    for (int i = 0; i < ITERS; ++i) {
        const long long b = lane + (i << 5);
        p[i] = pred[b * (long long)HW + (long long)(cc[i].x * W + cc[i].y)];
    }

    float partial = 0.0f;
#pragma unroll
    for (int i = 0; i < ITERS; ++i) {
        const float d = p[i] - tg[i];
        partial = fmaf(d, d, partial);
    }

    const float total = wave_reduce_wmma(partial);
    if (lane == 0)
        out[0] = total * inv_B;
}

// Generic fallback for B not a multiple-of-32 known shape (not expected here).
__global__ __launch_bounds__(32)
void center_mse_wmma_kernel_dyn(const float* __restrict__ pred,
                                const float* __restrict__ target,
                                const int2* __restrict__ c2,
                                float* __restrict__ out,
                                int B, int HW, int W, float inv_B)
{
    const int lane = (int)threadIdx.x;
    float partial = 0.0f;
    for (int b = lane; b < B; b += 32) {
        const int2 cc = c2[b];
        const float d = pred[(long long)b * HW + (long long)(cc.x * W + cc.y)]
                        - target[b];
        partial = fmaf(d, d, partial);
    }
    const float total = wave_reduce_wmma(partial);
    if (lane == 0)
        out[0] = total * inv_B;
}

extern "C" void kernel_launch(void* const* d_in, const int* in_sizes, int n_in,
                              void* d_out, int out_size, void* d_ws, size_t ws_size,
                              hipStream_t stream) {
    const float* pred    = (const float*)d_in[0];   // (B,1,H,W) f32
    const float* target  = (const float*)d_in[1];   // (B,)      f32
    const int2*  centers = (const int2*)d_in[2];    // (B,2)     i32
    float*       out     = (float*)d_out;           // scalar f32

    const int   B     = in_sizes[1];                // 512
    const int   HW    = in_sizes[0] / B;            // H*W = 65536
    const int   W     = 256;                        // reference: H = W = 256
    const float inv_B = 1.0f / (float)B;            // exact: B is a power of two

    if (B == 512) {
        center_mse_wmma_kernel<16><<<1, 32, 0, stream>>>(
            pred, target, centers, out, HW, W, inv_B);
    } else {
        center_mse_wmma_kernel_dyn<<<1, 32, 0, stream>>>(
            pred, target, centers, out, B, HW, W, inv_B);
    }
}